// SubstructureGNN_21191368638510
// MI455X (gfx1250) — compile-verified
//
#include <hip/hip_runtime.h>
#include <hip/hip_bf16.h>

// ---------------------------------------------------------------------------
// GAT (2-layer, 4 heads x 32 ch) for MI455X / gfx1250.
// GEMMs via V_WMMA_F32_16X16X4_F32 (fp32 matrix path, wave32).
// Edge phase: 3 passes with L2-resident gathers + coalesced f32 atomics.
// ---------------------------------------------------------------------------

typedef __attribute__((ext_vector_type(2))) float v2f;
typedef __attribute__((ext_vector_type(8))) float v8f;

#define FDIM 128            // Fin = HID = OUT = 128
#define NHEAD 4
#define CHAN 32             // channels per head
#define NEG_SLOPE 0.2f

// ---------------------------------------------------------------------------
// H = X (N x 128) @ W (128 x 128), fp32 WMMA 16x16x4.
// One wave per 16x16 output tile; 8 waves/block cover all 128 columns of a
// 16-row stripe. K loop = 32 WMMA ops.
// A layout (ISA 7.12.2, 32-bit A 16x4): lane m=lane&15, half=lane>>4;
//   a[0]=X[m][k+2*half], a[1]=X[m][k+2*half+1].
// B layout (rows striped across lanes): b[0]=W[k+2*half][col0+m], b[1]=next K.
// C/D layout: acc[v] -> row (v + 8*half), col (lane&15).
// ---------------------------------------------------------------------------
__global__ __launch_bounds__(256) void gat_gemm_wmma(const float* __restrict__ X,
                                                     const float* __restrict__ W,
                                                     float* __restrict__ H,
                                                     int N) {
  const int lane = threadIdx.x & 31;
  const int wave = threadIdx.x >> 5;       // 0..7 -> column tile
  const int m    = lane & 15;
  const int half = lane >> 4;              // 0 or 1
  const int row0 = blockIdx.x * 16;
  const int col0 = wave * 16;

  int rowA = row0 + m;
  if (rowA >= N) rowA = N - 1;             // clamp (N is a multiple of 16 here)
  const float* __restrict__ xrow = X + (size_t)rowA * FDIM;

  v8f acc = {};
#pragma unroll
  for (int k = 0; k < FDIM; k += 4) {
    const int ka = k + 2 * half;
    v2f a = *(const v2f*)(xrow + ka);                 // 8B aligned (ka even)
    v2f b;
    b.x = W[ka * FDIM + col0 + m];
    b.y = W[(ka + 1) * FDIM + col0 + m];
    acc = __builtin_amdgcn_wmma_f32_16x16x4_f32(
        /*neg_a=*/false, a, /*neg_b=*/false, b,
        /*c_mod=*/(short)0, acc, /*reuse_a=*/false, /*reuse_b=*/false);
  }

#pragma unroll
  for (int v = 0; v < 8; ++v) {
    const int row = row0 + v + 8 * half;
    if (row < N) H[(size_t)row * FDIM + col0 + m] = acc[v];
  }
}

// ---------------------------------------------------------------------------
// alpha_src[n,h] = dot(H[n, h*32 : h*32+32], a_src[h]);  same for alpha_dst.
// One thread per (node, head).
// ---------------------------------------------------------------------------
__global__ void gat_alpha(const float* __restrict__ H,
                          const float* __restrict__ a_src,
                          const float* __restrict__ a_dst,
                          float* __restrict__ as_, float* __restrict__ ad_,
                          int N) {
  const int tid = blockIdx.x * blockDim.x + threadIdx.x;
  if (tid >= N * NHEAD) return;
  const int n = tid >> 2;
  const int h = tid & 3;
  const float* __restrict__ hp = H + (size_t)n * FDIM + h * CHAN;
  const float* __restrict__ asv = a_src + h * CHAN;
  const float* __restrict__ adv = a_dst + h * CHAN;
  float s = 0.0f, d = 0.0f;
#pragma unroll
  for (int c = 0; c < CHAN; ++c) {
    const float hv = hp[c];
    s = fmaf(hv, asv[c], s);
    d = fmaf(hv, adv[c], d);
  }
  as_[tid] = s;
  ad_[tid] = d;
}

__global__ void fill_kernel(float* __restrict__ p, float v, int n) {
  const int i = blockIdx.x * blockDim.x + threadIdx.x;
  if (i < n) p[i] = v;
}

// Order-preserving float atomic max: signed-int max for v>=0, unsigned min for v<0.
__device__ __forceinline__ void atomicMaxFloat(float* addr, float v) {
  const int iv = __float_as_int(v);
  if (iv >= 0) atomicMax((int*)addr, iv);
  else         atomicMin((unsigned int*)addr, (unsigned int)iv);
}

__device__ __forceinline__ void edge_endpoints(const int* __restrict__ ei,
                                               int e, int E, int* s, int* d) {
  if (e < E) { *s = ei[e]; *d = ei[E + e]; }
  else       { *s = e - E; *d = e - E; }      // self loop
}

__device__ __forceinline__ float leaky(float v) {
  return v > 0.0f ? v : NEG_SLOPE * v;
}

// Pass 1: m[d,h] = max over incoming edges of leaky_relu(as[s,h]+ad[d,h])
__global__ void gat_edge_max(const int* __restrict__ ei, int E, int ET,
                             const float* __restrict__ as_,
                             const float* __restrict__ ad_,
                             float* __restrict__ m_) {
  const int tid = blockIdx.x * blockDim.x + threadIdx.x;
  if (tid >= ET * NHEAD) return;
  const int e = tid >> 2, h = tid & 3;
  int s, d;
  edge_endpoints(ei, e, E, &s, &d);
  const float v = leaky(as_[s * NHEAD + h] + ad_[d * NHEAD + h]);
  atomicMaxFloat(&m_[d * NHEAD + h], v);
}

// Pass 2: denom[d,h] += exp(e - m[d,h])
__global__ void gat_edge_expsum(const int* __restrict__ ei, int E, int ET,
                                const float* __restrict__ as_,
                                const float* __restrict__ ad_,
                                const float* __restrict__ m_,
                                float* __restrict__ den_) {
  const int tid = blockIdx.x * blockDim.x + threadIdx.x;
  if (tid >= ET * NHEAD) return;
  const int e = tid >> 2, h = tid & 3;
  int s, d;
  edge_endpoints(ei, e, E, &s, &d);
  const float v = leaky(as_[s * NHEAD + h] + ad_[d * NHEAD + h]);
  const float ex = __expf(v - m_[d * NHEAD + h]);
  atomicAdd(&den_[d * NHEAD + h], ex);
}

// Pass 3: out[d, :] += alpha(e,h) * H[s, :].
// One wave covers one edge's 128 channels: lane q handles 4 consecutive
// channels -> lane-consecutive atomic addresses (coalesced L2 atomics).
__global__ void gat_edge_scatter(const int* __restrict__ ei, int E, int ET,
                                 const float* __restrict__ as_,
                                 const float* __restrict__ ad_,
                                 const float* __restrict__ m_,
                                 const float* __restrict__ den_,
                                 const float* __restrict__ H,
                                 float* __restrict__ out) {
  const long long tid = (long long)blockIdx.x * blockDim.x + threadIdx.x;
  if (tid >= (long long)ET * 32) return;
  const int e  = (int)(tid >> 5);
  const int q  = (int)(tid & 31);
  const int c0 = q * 4;                 // 0..124
  const int h  = c0 >> 5;               // head of this 4-channel chunk
  int s, d;
  edge_endpoints(ei, e, E, &s, &d);
  const float v  = leaky(as_[s * NHEAD + h] + ad_[d * NHEAD + h]);
  const float ex = __expf(v - m_[d * NHEAD + h]);
  const float alpha = ex / (den_[d * NHEAD + h] + 1e-16f);

  const float4 hv = *(const float4*)(H + (size_t)s * FDIM + c0);
  float* __restrict__ op = out + (size_t)d * FDIM + c0;
  atomicAdd(op + 0, alpha * hv.x);
  atomicAdd(op + 1, alpha * hv.y);
  atomicAdd(op + 2, alpha * hv.z);
  atomicAdd(op + 3, alpha * hv.w);
}

// out[i] = (accum[i] + b[i % 128]), optional ReLU (layer 1).
__global__ void gat_bias_act(float* __restrict__ p, const float* __restrict__ b,
                             int total, int do_relu) {
  const int i = blockIdx.x * blockDim.x + threadIdx.x;
  if (i >= total) return;
  float v = p[i] + b[i & (FDIM - 1)];
  if (do_relu) v = v > 0.0f ? v : 0.0f;
  p[i] = v;
}

// ---------------------------------------------------------------------------
static inline int cdiv(long long a, int b) { return (int)((a + b - 1) / b); }

extern "C" void kernel_launch(void* const* d_in, const int* in_sizes, int n_in,
                              void* d_out, int out_size, void* d_ws, size_t ws_size,
                              hipStream_t stream) {
  const float* x     = (const float*)d_in[0];
  const int*   ei    = (const int*)d_in[1];
  const float* W1    = (const float*)d_in[2];
  const float* asrc1 = (const float*)d_in[3];
  const float* adst1 = (const float*)d_in[4];
  const float* b1    = (const float*)d_in[5];
  const float* W2    = (const float*)d_in[6];
  const float* asrc2 = (const float*)d_in[7];
  const float* adst2 = (const float*)d_in[8];
  const float* b2    = (const float*)d_in[9];
  float* out = (float*)d_out;

  const int N  = in_sizes[0] / FDIM;       // 50000
  const int E  = in_sizes[1] / 2;          // 800000
  const int ET = E + N;                    // + self loops

  // Workspace carve (floats): H, Hact: N*128 each; 4 small N*4 arrays. ~54 MB.
  float* H    = (float*)d_ws;
  float* Hact = H    + (size_t)N * FDIM;
  float* as_  = Hact + (size_t)N * FDIM;
  float* ad_  = as_  + (size_t)N * NHEAD;
  float* m_   = ad_  + (size_t)N * NHEAD;
  float* den_ = m_   + (size_t)N * NHEAD;

  const int TB = 256;
  const int gTiles = cdiv(N, 16);
  const int gNH    = cdiv((long long)N * NHEAD, TB);
  const int gNF    = cdiv((long long)N * FDIM, TB);
  const int gEH    = cdiv((long long)ET * NHEAD, TB);
  const int gE32   = cdiv((long long)ET * 32, TB);
  const float NEG_INF = -__builtin_huge_valf();

  // ------------------------- Layer 1 -------------------------
  fill_kernel<<<gNH, TB, 0, stream>>>(m_, NEG_INF, N * NHEAD);
  fill_kernel<<<gNH, TB, 0, stream>>>(den_, 0.0f, N * NHEAD);
  fill_kernel<<<gNF, TB, 0, stream>>>(Hact, 0.0f, N * FDIM);

  gat_gemm_wmma<<<gTiles, TB, 0, stream>>>(x, W1, H, N);
  gat_alpha<<<gNH, TB, 0, stream>>>(H, asrc1, adst1, as_, ad_, N);
  gat_edge_max<<<gEH, TB, 0, stream>>>(ei, E, ET, as_, ad_, m_);
  gat_edge_expsum<<<gEH, TB, 0, stream>>>(ei, E, ET, as_, ad_, m_, den_);
  gat_edge_scatter<<<gE32, TB, 0, stream>>>(ei, E, ET, as_, ad_, m_, den_, H, Hact);
  gat_bias_act<<<gNF, TB, 0, stream>>>(Hact, b1, N * FDIM, /*relu=*/1);

  // ------------------------- Layer 2 -------------------------
  fill_kernel<<<gNH, TB, 0, stream>>>(m_, NEG_INF, N * NHEAD);
  fill_kernel<<<gNH, TB, 0, stream>>>(den_, 0.0f, N * NHEAD);
  fill_kernel<<<gNF, TB, 0, stream>>>(out, 0.0f, N * FDIM);

  gat_gemm_wmma<<<gTiles, TB, 0, stream>>>(Hact, W2, H, N);   // H reused
  gat_alpha<<<gNH, TB, 0, stream>>>(H, asrc2, adst2, as_, ad_, N);
  gat_edge_max<<<gEH, TB, 0, stream>>>(ei, E, ET, as_, ad_, m_);
  gat_edge_expsum<<<gEH, TB, 0, stream>>>(ei, E, ET, as_, ad_, m_, den_);
  gat_edge_scatter<<<gE32, TB, 0, stream>>>(ei, E, ET, as_, ad_, m_, den_, H, out);
  gat_bias_act<<<gNF, TB, 0, stream>>>(out, b2, N * FDIM, /*relu=*/0);
}